// SHWarpBlock_10977936408671
// MI455X (gfx1250) — compile-verified
//
#include <hip/hip_runtime.h>

// ---- problem constants (B=1, IN=2, C=16) ----
#define V_N   40962
#define V16   40976          // V padded to multiple of 16
#define F_N   81920          // faces (multiple of 16)
#define NH_N  6561           // (L+1)^2, L=80
#define NHP   6564           // NH padded to multiple of 4
#define ND_N  1681           // (DL+1)^2, DL=40
#define NDP   1684           // ND padded to multiple of 4
#define VFLOOR 40960         // largest multiple of 4 <= V_N (tail = 2 cols)
#define NTILE_T (V16 / 16)   // 2561 vertex tiles

typedef float v2f __attribute__((ext_vector_type(2)));
typedef float v8f __attribute__((ext_vector_type(8)));
typedef unsigned long long u64;

// ---- workspace layout (float offsets) ----
constexpr size_t OFF_R    = 0;                                 // 16
constexpr size_t OFF_CTRX = 16;
constexpr size_t OFF_CTRY = OFF_CTRX + F_N;
constexpr size_t OFF_CTRZ = OFF_CTRY + F_N;
constexpr size_t OFF_C2   = OFF_CTRZ + F_N;
constexpr size_t OFF_QX   = OFF_C2   + F_N;
constexpr size_t OFF_QY   = OFF_QX   + V16;
constexpr size_t OFF_QZ   = OFF_QY   + V16;
constexpr size_t OFF_FIDP = OFF_QZ   + V16;                    // u64[V16] packed (dist|idx)
constexpr size_t OFF_HA   = OFF_FIDP + 2 * (size_t)V16;        // 16 x V16
constexpr size_t OFF_CF   = OFF_HA   + 16 * (size_t)V16;       // 16 x NH
constexpr size_t OFF_CFW  = OFF_CF   + 16 * (size_t)NH_N;      // 16 x NHP
constexpr size_t OFF_H2   = OFF_CFW  + 16 * (size_t)NHP;       // 16 x V16 (pre-ReLU accum)
constexpr size_t OFF_XLA  = OFF_H2   + 16 * (size_t)V16;       // 16 x V16
constexpr size_t OFF_COEF = OFF_XLA  + 16 * (size_t)V16;       // 16 x ND
constexpr size_t OFF_CW   = OFF_COEF + 16 * (size_t)ND_N;      // 16 x NDP
constexpr size_t OFF_X3   = OFF_CW   + 16 * (size_t)NDP;       // 6 x V16 accum

__device__ __forceinline__ void rot6d_dev(const float x[6], float R[9]) {
    float a1x = x[0], a1y = x[1], a1z = x[2];
    float n1 = sqrtf(a1x * a1x + a1y * a1y + a1z * a1z) + 1e-8f;
    float r1x = a1x / n1, r1y = a1y / n1, r1z = a1z / n1;
    float a2x = x[3], a2y = x[4], a2z = x[5];
    float dp = r1x * a2x + r1y * a2y + r1z * a2z;
    a2x -= dp * r1x; a2y -= dp * r1y; a2z -= dp * r1z;
    float n2 = sqrtf(a2x * a2x + a2y * a2y + a2z * a2z) + 1e-8f;
    float r2x = a2x / n2, r2y = a2y / n2, r2z = a2z / n2;
    R[0] = r1x; R[1] = r1y; R[2] = r1z;
    R[3] = r2x; R[4] = r2y; R[5] = r2z;
    R[6] = r1y * r2z - r1z * r2y;
    R[7] = r1z * r2x - r1x * r2z;
    R[8] = r1x * r2y - r1y * r2x;
}

// K0: global weighted reduction -> rot6d -> global rotation R (9 floats)
__global__ void k_rotinit(const float* __restrict__ inp, const float* __restrict__ w0,
                          const float* __restrict__ W0, const float* __restrict__ b0,
                          float* __restrict__ Rout) {
    __shared__ float s0[256], s1[256];
    int tid = threadIdx.x;
    float a0 = 0.f, a1 = 0.f;
    for (int t = tid; t < V_N; t += 256) {
        float aw = fabsf(w0[t]);
        a0 += inp[t] * aw;
        a1 += inp[V_N + t] * aw;
    }
    s0[tid] = a0; s1[tid] = a1;
    __syncthreads();
    for (int st = 128; st > 0; st >>= 1) {
        if (tid < st) { s0[tid] += s0[tid + st]; s1[tid] += s1[tid + st]; }
        __syncthreads();
    }
    if (tid == 0) {
        float x0 = s0[0] * 0.001f, x1 = s1[0] * 0.001f;
        float x6[6];
        for (int o = 0; o < 6; ++o) x6[o] = W0[o * 2] * x0 + W0[o * 2 + 1] * x1 + b0[o];
        float R[9];
        rot6d_dev(x6, R);
        for (int i = 0; i < 9; ++i) Rout[i] = R[i];
    }
}

// K1: face centroids (SoA) + |ctr|^2
__global__ void k_centroid(const int* __restrict__ f, const float* __restrict__ v,
                           float* __restrict__ cx, float* __restrict__ cy,
                           float* __restrict__ cz, float* __restrict__ c2) {
    int t = blockIdx.x * blockDim.x + threadIdx.x;
    if (t >= F_N) return;
    int i0 = f[t * 3], i1 = f[t * 3 + 1], i2 = f[t * 3 + 2];
    float x = (v[i0 * 3 + 0] + v[i1 * 3 + 0] + v[i2 * 3 + 0]) * (1.f / 3.f);
    float y = (v[i0 * 3 + 1] + v[i1 * 3 + 1] + v[i2 * 3 + 1]) * (1.f / 3.f);
    float z = (v[i0 * 3 + 2] + v[i1 * 3 + 2] + v[i2 * 3 + 2]) * (1.f / 3.f);
    cx[t] = x; cy[t] = y; cz[t] = z;
    c2[t] = x * x + y * y + z * z;
}

// K2: q = R * v  (SoA, zero-padded to V16)
__global__ void k_rotate(const float* __restrict__ R, const float* __restrict__ v,
                         float* __restrict__ qx, float* __restrict__ qy, float* __restrict__ qz) {
    int t = blockIdx.x * blockDim.x + threadIdx.x;
    if (t >= V16) return;
    if (t >= V_N) { qx[t] = 0.f; qy[t] = 0.f; qz[t] = 0.f; return; }
    float vx = v[t * 3], vy = v[t * 3 + 1], vz = v[t * 3 + 2];
    qx[t] = R[0] * vx + R[1] * vy + R[2] * vz;
    qy[t] = R[3] * vx + R[4] * vy + R[5] * vz;
    qz[t] = R[6] * vx + R[7] * vy + R[8] * vz;
}

// K3: nearest face via WMMA f32 16x16x4.
// A row m = (ctr_x, ctr_y, ctr_z, |ctr|^2), B col n = (-2qx, -2qy, -2qz, 1)
// => D[m][n] = |ctr_m|^2 - 2 q_n . ctr_m.
// 8 parallel (min, tile) slots per lane; 2x manual unroll with pointer-bump
// addressing; gridDim.y = 8 face-chunks merged via atomicMin on (dist|idx) u64
// key (reproduces argmin's lowest-index tie-break).
__global__ void k_nearest(const float* __restrict__ qx, const float* __restrict__ qy,
                          const float* __restrict__ qz, const float* __restrict__ cx,
                          const float* __restrict__ cy, const float* __restrict__ cz,
                          const float* __restrict__ c2, u64* __restrict__ fidp) {
    int wid = (blockIdx.x * blockDim.x + threadIdx.x) >> 5;
    if (wid >= NTILE_T) return;                     // wave-uniform guard
    int lane = threadIdx.x & 31;
    int m = lane & 15;
    bool lo = lane < 16;
    int qb = wid * 16;
    v2f B;
    if (lo) { B[0] = -2.0f * qx[qb + m]; B[1] = -2.0f * qy[qb + m]; }
    else    { B[0] = -2.0f * qz[qb + m]; B[1] = 1.0f; }
    int rbase = lo ? 0 : 8;

    float best[8];
    int   bft[8];
#pragma unroll
    for (int r = 0; r < 8; ++r) { best[r] = 3.4e38f; bft[r] = 0; }

    const int tilesPerChunk = (F_N / 16) / 8;       // 640 (even)
    int ft0 = blockIdx.y * tilesPerChunk;
    const float* p0 = (lo ? cx : cz) + (size_t)ft0 * 16 + m;
    const float* p1 = (lo ? cy : c2) + (size_t)ft0 * 16 + m;
    for (int ft = ft0; ft < ft0 + tilesPerChunk; ft += 2) {
        v2f A0; A0[0] = p0[0];  A0[1] = p1[0];
        v2f A1; A1[0] = p0[16]; A1[1] = p1[16];
        p0 += 32; p1 += 32;
        v8f d0 = {};
        d0 = __builtin_amdgcn_wmma_f32_16x16x4_f32(false, A0, false, B, (short)0, d0, false, false);
        v8f d1 = {};
        d1 = __builtin_amdgcn_wmma_f32_16x16x4_f32(false, A1, false, B, (short)0, d1, false, false);
#pragma unroll
        for (int r = 0; r < 8; ++r) {               // 8 independent chains
            if (d0[r] < best[r]) { best[r] = d0[r]; bft[r] = ft; }
        }
#pragma unroll
        for (int r = 0; r < 8; ++r) {
            if (d1[r] < best[r]) { best[r] = d1[r]; bft[r] = ft + 1; }
        }
    }
    // merge the 8 slots (scan order ~ original)
    float gb = best[0];
    int   gi = bft[0] * 16 + rbase;
#pragma unroll
    for (int r = 1; r < 8; ++r) {
        if (best[r] < gb) { gb = best[r]; gi = bft[r] * 16 + rbase + r; }
    }
    // pack order-preserving float key | face index
    unsigned ub = __float_as_uint(gb);
    ub = (ub & 0x80000000u) ? ~ub : (ub | 0x80000000u);
    u64 pack = ((u64)ub << 32) | (unsigned)gi;
    // merge the two lane halves (rows 0-7 vs 8-15 of each column)
    u64 opack = __shfl_xor(pack, 16, 32);
    if (opack < pack) pack = opack;
    int q = qb + m;
    if (lo && q < V_N) atomicMin(&fidp[q], pack);
}

// K4: barycentric weights + input gather + Wv1 MLP + *area -> ha (16 x V16, zero pad)
__global__ void k_gather(const u64* __restrict__ fidp, const int* __restrict__ f,
                         const float* __restrict__ v, const float* __restrict__ qx,
                         const float* __restrict__ qy, const float* __restrict__ qz,
                         const float* __restrict__ inp, const float* __restrict__ Wv1,
                         const float* __restrict__ bv1, const float* __restrict__ area,
                         float* __restrict__ ha) {
    int t = blockIdx.x * blockDim.x + threadIdx.x;
    if (t >= V16) return;
    if (t >= V_N) {
        for (int c = 0; c < 16; ++c) ha[(size_t)c * V16 + t] = 0.f;
        return;
    }
    int fi = (int)(unsigned)(fidp[t] & 0xFFFFFFFFull);
    int i0 = f[fi * 3], i1 = f[fi * 3 + 1], i2 = f[fi * 3 + 2];
    float ax = v[i0 * 3], ay = v[i0 * 3 + 1], az = v[i0 * 3 + 2];
    float bx = v[i1 * 3], by = v[i1 * 3 + 1], bz = v[i1 * 3 + 2];
    float ccx = v[i2 * 3], ccy = v[i2 * 3 + 1], ccz = v[i2 * 3 + 2];
    float e0x = bx - ax, e0y = by - ay, e0z = bz - az;
    float e1x = ccx - ax, e1y = ccy - ay, e1z = ccz - az;
    float e2x = qx[t] - ax, e2y = qy[t] - ay, e2z = qz[t] - az;
    float d00 = e0x * e0x + e0y * e0y + e0z * e0z;
    float d01 = e0x * e1x + e0y * e1y + e0z * e1z;
    float d11 = e1x * e1x + e1y * e1y + e1z * e1z;
    float d20 = e2x * e0x + e2y * e0y + e2z * e0z;
    float d21 = e2x * e1x + e2y * e1y + e2z * e1z;
    float den = d00 * d11 - d01 * d01 + 1e-12f;
    float w1 = (d11 * d20 - d01 * d21) / den;
    float w2 = (d00 * d21 - d01 * d20) / den;
    float w0b = 1.f - w1 - w2;
    float x0 = inp[i0] * w0b + inp[i1] * w1 + inp[i2] * w2;
    float x1 = inp[V_N + i0] * w0b + inp[V_N + i1] * w1 + inp[V_N + i2] * w2;
    float ar = area[t];
    for (int c = 0; c < 16; ++c) {
        float h = Wv1[c * 2] * x0 + Wv1[c * 2 + 1] * x1 + bv1[c];
        ha[(size_t)c * V16 + t] = fmaxf(h, 0.f) * ar;
    }
}

// K5/K8: out[c,n] += sum_v Aw[c,v] * Y[n,v]  (M=16, K=V, N=NR), WMMA 16x16x4 f32.
// 16-way K-split: blockIdx.x = N-tile, kc = blockIdx.y*8 + warp; f32 atomics merge.
__global__ void k_gemm_AYt(const float* __restrict__ Aw, const float* __restrict__ Y,
                           float* __restrict__ out, int NR) {
    int lane = threadIdx.x & 31;
    int kc = blockIdx.y * 8 + (threadIdx.x >> 5);   // 0..15
    int m = lane & 15;
    bool lo = lane < 16;
    int koff = lo ? 0 : 2;
    int nb = blockIdx.x * 16;
    int nrow = nb + m; if (nrow >= NR) nrow = NR - 1;     // clamp keeps loads in-bounds
    const float* yro = Y + (size_t)nrow * V_N + koff;
    const float* aro = Aw + (size_t)m * V16 + koff;
    int start = kc * (VFLOOR / 16);                 // 2560, multiple of 4
    int end = start + (VFLOOR / 16);
    v8f d = {};
    for (int vv = start; vv < end; vv += 4) {
        v2f a = *(const v2f*)(aro + vv);
        v2f b = *(const v2f*)(yro + vv);
        d = __builtin_amdgcn_wmma_f32_16x16x4_f32(false, a, false, b, (short)0, d, false, false);
    }
    if (kc == 15) {                                 // scalar tail: v = V_N-2, V_N-1
        float y0 = Y[(size_t)nrow * V_N + V_N - 2];
        float y1 = Y[(size_t)nrow * V_N + V_N - 1];
#pragma unroll
        for (int r = 0; r < 8; ++r) {
            int c = (lo ? 0 : 8) + r;
            d[r] += Aw[(size_t)c * V16 + V_N - 2] * y0 + Aw[(size_t)c * V16 + V_N - 1] * y1;
        }
    }
    int n = nb + m;
    if (n < NR) {
#pragma unroll
        for (int r = 0; r < 8; ++r) {
            int c = (lo ? 0 : 8) + r;
            atomicAdd(&out[(size_t)c * NR + n], d[r]);
        }
    }
}

// K5b: cfw = cf * wspec, zero-padded to 16 x NHP
__global__ void k_cfw(const float* __restrict__ cf, const float* __restrict__ wspec,
                      float* __restrict__ cfw) {
    int i = blockIdx.x * blockDim.x + threadIdx.x;
    if (i >= 16 * NHP) return;
    int c = i / NHP, n = i % NHP;
    float val = 0.f;
    if (n < NH_N) val = cf[(size_t)c * NH_N + n] * wspec[(size_t)c * NH_N + n];
    cfw[i] = val;
}

// K6: h2[c,t] += sum_k cfw[c,k] * Y[k,t]  (K=NH, N=V), WMMA, 8-way K-split (gridDim.y),
// pre-ReLU accumulation via f32 atomics (ReLU applied in k_vertmlp).
__global__ void k_specback(const float* __restrict__ cfw, const float* __restrict__ Y,
                           float* __restrict__ h2) {
    int wid = (blockIdx.x * blockDim.x + threadIdx.x) >> 5;
    if (wid >= NTILE_T) return;
    int lane = threadIdx.x & 31;
    int m = lane & 15;
    bool lo = lane < 16;
    int koff = lo ? 0 : 2;
    int tb = wid * 16;
    int t = tb + m; if (t >= V_N) t = V_N - 1;      // clamp for B loads
    const float* aro = cfw + (size_t)m * NHP + koff;
    int k0 = blockIdx.y * 820;                      // 6560/8, multiple of 4
    int k1 = k0 + 820;
    v8f d = {};
    for (int k = k0; k < k1; k += 4) {
        v2f a = *(const v2f*)(aro + k);
        v2f b;
        b[0] = Y[(size_t)(k + koff) * V_N + t];
        b[1] = Y[(size_t)(k + koff + 1) * V_N + t];
        d = __builtin_amdgcn_wmma_f32_16x16x4_f32(false, a, false, b, (short)0, d, false, false);
    }
    if (blockIdx.y == 7) {                          // tail row k = 6560
        float yk = Y[(size_t)6560 * V_N + t];
#pragma unroll
        for (int r = 0; r < 8; ++r) {
            int c = (lo ? 0 : 8) + r;
            d[r] += cfw[(size_t)c * NHP + 6560] * yk;
        }
    }
    int tt = tb + m;
    if (tt < V_N) {
#pragma unroll
        for (int r = 0; r < 8; ++r) {
            int c = (lo ? 0 : 8) + r;
            atomicAdd(&h2[(size_t)c * V16 + tt], d[r]);
        }
    }
}

// K7: per-vertex 16x16 MLP (ReLU(h2), Wv2, relu) + Wl + *area -> xla (rows 6..15 zero)
__global__ void k_vertmlp(const float* __restrict__ h2, const float* __restrict__ Wv2,
                          const float* __restrict__ bv2, const float* __restrict__ Wl,
                          const float* __restrict__ bl, const float* __restrict__ area,
                          float* __restrict__ xla) {
    int t = blockIdx.x * blockDim.x + threadIdx.x;
    if (t >= V16) return;
    if (t >= V_N) {
        for (int c = 0; c < 16; ++c) xla[(size_t)c * V16 + t] = 0.f;
        return;
    }
    float h[16];
    for (int c = 0; c < 16; ++c) h[c] = fmaxf(h2[(size_t)c * V16 + t], 0.f);
    float x2[16];
    for (int o = 0; o < 16; ++o) {
        float s = bv2[o];
        for (int i = 0; i < 16; ++i) s += Wv2[o * 16 + i] * h[i];
        x2[o] = fmaxf(s, 0.f);
    }
    float ar = area[t];
    for (int o = 0; o < 6; ++o) {
        float s = bl[o];
        for (int i = 0; i < 16; ++i) s += Wl[o * 16 + i] * x2[i];
        xla[(size_t)o * V16 + t] = s * ar;
    }
    for (int o = 6; o < 16; ++o) xla[(size_t)o * V16 + t] = 0.f;
}

// K9: cw = coef*w + b, padded to 16 x NDP (rows >=6 and cols >= ND are zero)
__global__ void k_cw(const float* __restrict__ coef, const float* __restrict__ w,
                     const float* __restrict__ b, float* __restrict__ cw) {
    int i = blockIdx.x * blockDim.x + threadIdx.x;
    if (i >= 16 * NDP) return;
    int c = i / NDP, n = i % NDP;
    float val = 0.f;
    if (c < 6 && n < ND_N) val = coef[(size_t)c * ND_N + n] * w[c * ND_N + n] + b[c * ND_N + n];
    cw[i] = val;
}

// K10: x3[c,t] += sum_k cw[c,k]*Y[k,t] (K=NDP; Y rows 0..NDP-1 all exist since NDP<NH),
// 4-way K-split (gridDim.y); only rows c=0..5 are non-zero -> atomics only for those.
__global__ void k_final(const float* __restrict__ cw, const float* __restrict__ Y,
                        float* __restrict__ x3) {
    int wid = (blockIdx.x * blockDim.x + threadIdx.x) >> 5;
    if (wid >= NTILE_T) return;
    int lane = threadIdx.x & 31;
    int m = lane & 15;
    bool lo = lane < 16;
    int koff = lo ? 0 : 2;
    int tb = wid * 16;
    int t = tb + m; if (t >= V_N) t = V_N - 1;
    const float* aro = cw + (size_t)m * NDP + koff;
    int k0 = blockIdx.y * 424;                      // multiple of 4
    int k1 = k0 + 424; if (k1 > NDP) k1 = NDP;
    v8f d = {};
    for (int k = k0; k < k1; k += 4) {
        v2f a = *(const v2f*)(aro + k);
        v2f b;
        b[0] = Y[(size_t)(k + koff) * V_N + t];
        b[1] = Y[(size_t)(k + koff + 1) * V_N + t];
        d = __builtin_amdgcn_wmma_f32_16x16x4_f32(false, a, false, b, (short)0, d, false, false);
    }
    int tt = tb + m;
    if (lo && tt < V_N) {                           // lanes 0-15 hold rows c=0..7
#pragma unroll
        for (int r = 0; r < 6; ++r) atomicAdd(&x3[(size_t)r * V16 + tt], d[r]);
    }
}

// K11: per-vertex rot6d + rotate v -> out (V x 3)
__global__ void k_out(const float* __restrict__ x3, const float* __restrict__ v,
                      float* __restrict__ out) {
    int t = blockIdx.x * blockDim.x + threadIdx.x;
    if (t >= V_N) return;
    float x6[6];
    for (int i = 0; i < 6; ++i) x6[i] = x3[(size_t)i * V16 + t];
    float R[9];
    rot6d_dev(x6, R);
    float vx = v[t * 3], vy = v[t * 3 + 1], vz = v[t * 3 + 2];
    out[t * 3 + 0] = R[0] * vx + R[1] * vy + R[2] * vz;
    out[t * 3 + 1] = R[3] * vx + R[4] * vy + R[5] * vz;
    out[t * 3 + 2] = R[6] * vx + R[7] * vy + R[8] * vz;
}

extern "C" void kernel_launch(void* const* d_in, const int* in_sizes, int n_in,
                              void* d_out, int out_size, void* d_ws, size_t ws_size,
                              hipStream_t stream) {
    const float* inp   = (const float*)d_in[0];
    const float* v     = (const float*)d_in[1];
    const int*   f     = (const int*)  d_in[2];
    const float* area  = (const float*)d_in[3];
    const float* Y     = (const float*)d_in[4];
    const float* w0    = (const float*)d_in[5];
    const float* W0    = (const float*)d_in[6];
    const float* b0    = (const float*)d_in[7];
    const float* Wv1   = (const float*)d_in[8];
    const float* bv1   = (const float*)d_in[9];
    const float* wspec = (const float*)d_in[10];
    const float* Wv2   = (const float*)d_in[11];
    const float* bv2   = (const float*)d_in[12];
    const float* Wl    = (const float*)d_in[13];
    const float* bl    = (const float*)d_in[14];
    const float* w     = (const float*)d_in[15];
    const float* b     = (const float*)d_in[16];
    float* out = (float*)d_out;

    float* ws   = (float*)d_ws;
    float* R    = ws + OFF_R;
    float* ctrx = ws + OFF_CTRX;
    float* ctry = ws + OFF_CTRY;
    float* ctrz = ws + OFF_CTRZ;
    float* c2   = ws + OFF_C2;
    float* qx   = ws + OFF_QX;
    float* qy   = ws + OFF_QY;
    float* qz   = ws + OFF_QZ;
    u64*   fidp = (u64*)(ws + OFF_FIDP);
    float* ha   = ws + OFF_HA;
    float* cf   = ws + OFF_CF;
    float* cfw  = ws + OFF_CFW;
    float* h2   = ws + OFF_H2;
    float* xla  = ws + OFF_XLA;
    float* coef = ws + OFF_COEF;
    float* cw   = ws + OFF_CW;
    float* x3   = ws + OFF_X3;

    const int vblocks = (V16 + 255) / 256;          // 161
    const int wblocks = (NTILE_T + 7) / 8;          // 321 (8 waves/block)

    k_rotinit <<<1, 256, 0, stream>>>(inp, w0, W0, b0, R);
    k_centroid<<<F_N / 256, 256, 0, stream>>>(f, v, ctrx, ctry, ctrz, c2);
    k_rotate  <<<vblocks, 256, 0, stream>>>(R, v, qx, qy, qz);

    hipMemsetAsync(fidp, 0xFF, sizeof(u64) * (size_t)V16, stream);
    k_nearest <<<dim3(wblocks, 8), 256, 0, stream>>>(qx, qy, qz, ctrx, ctry, ctrz, c2, fidp);
    k_gather  <<<vblocks, 256, 0, stream>>>(fidp, f, v, qx, qy, qz, inp, Wv1, bv1, area, ha);

    hipMemsetAsync(cf, 0, sizeof(float) * 16 * (size_t)NH_N, stream);
    k_gemm_AYt<<<dim3((NH_N + 15) / 16, 2), 256, 0, stream>>>(ha, Y, cf, NH_N);
    k_cfw     <<<(16 * NHP + 255) / 256, 256, 0, stream>>>(cf, wspec, cfw);

    hipMemsetAsync(h2, 0, sizeof(float) * 16 * (size_t)V16, stream);
    k_specback<<<dim3(wblocks, 8), 256, 0, stream>>>(cfw, Y, h2);
    k_vertmlp <<<vblocks, 256, 0, stream>>>(h2, Wv2, bv2, Wl, bl, area, xla);

    hipMemsetAsync(coef, 0, sizeof(float) * 16 * (size_t)ND_N, stream);
    k_gemm_AYt<<<dim3((ND_N + 15) / 16, 2), 256, 0, stream>>>(xla, Y, coef, ND_N);
    k_cw      <<<(16 * NDP + 255) / 256, 256, 0, stream>>>(coef, w, b, cw);

    hipMemsetAsync(x3, 0, sizeof(float) * 6 * (size_t)V16, stream);
    k_final   <<<dim3(wblocks, 4), 256, 0, stream>>>(cw, Y, x3);
    k_out     <<<(V_N + 255) / 256, 256, 0, stream>>>(x3, v, out);
}